// TargetModel_78786880077968
// MI455X (gfx1250) — compile-verified
//
#include <hip/hip_runtime.h>
#include <hip/hip_bf16.h>

#define N_NODES_C  100000
#define N_EDGES_C  1600000
#define N_GRAPHS_C 512
#define DIM        64
#define OUTD       2

typedef __attribute__((ext_vector_type(2))) float v2f;
typedef __attribute__((ext_vector_type(8))) float v8f;

// ---------------- helpers ----------------
__global__ void k_zero_f32(float* __restrict__ p, int n) {
    int i = blockIdx.x * blockDim.x + threadIdx.x;
    if (i < n) p[i] = 0.0f;
}

__global__ void k_init_deg(float* __restrict__ deg) {
    int i = blockIdx.x * blockDim.x + threadIdx.x;
    if (i < N_NODES_C) deg[i] = 1.0f;           // self-loop contribution
}

__global__ void k_accum_deg(const int* __restrict__ dst, float* __restrict__ deg) {
    int e = blockIdx.x * blockDim.x + threadIdx.x;
    if (e < N_EDGES_C) atomicAdd(&deg[dst[e]], 1.0f);
}

__global__ void k_deg_to_dinv(float* __restrict__ deg) {
    int i = blockIdx.x * blockDim.x + threadIdx.x;
    if (i < N_NODES_C) deg[i] = rsqrtf(deg[i]); // deg >= 1 always (self-loops)
}

// ---------------- GEMM: H = X (nrows x 64) @ W (64 x 64), fp32 WMMA ----------------
// Block: 128 threads = 4 waves; each wave computes a 16x64 output slab.
// W is staged in LDS as K-pairs so each lane's B operand (W[k][n], W[k+1][n])
// is one contiguous 8-byte word -> single ds_load_b64, no register shuffles.
__global__ __launch_bounds__(128) void k_gemm64_wmma(const float* __restrict__ X,
                                                     const float* __restrict__ W,
                                                     float* __restrict__ H,
                                                     int nrows) {
    __shared__ v2f lw2[(DIM / 2) * DIM];        // [kpair][n], 2048 * 8B = 16 KB
    const int tid = threadIdx.x;

    // cooperative paired load of W: entry idx -> kpair = idx>>6, n = idx&63
    for (int idx = tid; idx < (DIM / 2) * DIM; idx += 128) {
        const int kp = idx >> 6;
        const int n  = idx & 63;
        v2f w;
        w.x = W[(2 * kp    ) * DIM + n];
        w.y = W[(2 * kp + 1) * DIM + n];
        lw2[idx] = w;
    }
    __syncthreads();

    const int wave = tid >> 5;
    const int lane = tid & 31;
    const int rowbase = blockIdx.x * 64 + wave * 16;
    const int m  = lane & 15;
    const int kb = (lane >> 4) * 2;             // ISA A-layout: lanes 0-15 -> K{0,1}, 16-31 -> K{2,3}
    const int nl = lane & 15;
    int row = rowbase + m;
    const int rl = row < nrows ? row : nrows - 1;   // clamp: keep EXEC all-ones for WMMA

    // preload A operands: 16 K-steps, stride-4 float pairs per ISA layout
    v2f a[16];
    const float* xr = X + (size_t)rl * DIM + kb;
    #pragma unroll
    for (int i = 0; i < 16; ++i) {
        a[i].x = xr[4 * i + 0];
        a[i].y = xr[4 * i + 1];
    }

    v8f acc[4];
    #pragma unroll
    for (int n = 0; n < 4; ++n) acc[n] = (v8f){0.f,0.f,0.f,0.f,0.f,0.f,0.f,0.f};

    // per-lane LDS base in paired units: kpair_lane = lane>>4, column = nl
    const v2f* lb = lw2 + (lane >> 4) * DIM + nl;
    #pragma unroll
    for (int i = 0; i < 16; ++i) {
        #pragma unroll
        for (int n = 0; n < 4; ++n) {
            const v2f b = lb[2 * i * DIM + n * 16];  // ds offset: i*1024 + n*128 bytes
            // D = A(16x4,f32) x B(4x16,f32) + C(16x16,f32)
            acc[n] = __builtin_amdgcn_wmma_f32_16x16x4_f32(
                         false, a[i], false, b, (short)0, acc[n], false, false);
        }
    }

    // D layout: VGPR v, lanes 0-15: M=v, N=lane; lanes 16-31: M=v+8, N=lane-16
    const int rstore = rowbase + 8 * (lane >> 4);
    #pragma unroll
    for (int v = 0; v < 8; ++v) {
        const int r = rstore + v;
        if (r < nrows) {
            #pragma unroll
            for (int n = 0; n < 4; ++n)
                H[(size_t)r * DIM + n * 16 + nl] = acc[n][v];
        }
    }
}

// ---------------- edge scatter: agg[dst] += h[src] * dinv[src]*dinv[dst] ----------------
// 16 lanes per edge, 4 features each (float4 gather, scalar fp32 atomics)
__global__ void k_scatter_edges(const float* __restrict__ H,
                                const int* __restrict__ src,
                                const int* __restrict__ dst,
                                const float* __restrict__ dinv,
                                float* __restrict__ agg) {
    long long gid = (long long)blockIdx.x * blockDim.x + threadIdx.x;
    const long long total = (long long)N_EDGES_C * 16;
    if (gid >= total) return;
    const int e = (int)(gid >> 4);
    const int f = ((int)gid & 15) * 4;
    const int s = src[e];
    const int d = dst[e];
    const float norm = dinv[s] * dinv[d];
    const float4 hv = *(const float4*)&H[(size_t)s * DIM + f];
    float* ap = &agg[(size_t)d * DIM + f];
    atomicAdd(ap + 0, hv.x * norm);
    atomicAdd(ap + 1, hv.y * norm);
    atomicAdd(ap + 2, hv.z * norm);
    atomicAdd(ap + 3, hv.w * norm);
}

// ---------------- finalize: agg = relu(agg + h*dinv^2 + bias)  (self-loop folded in) ----------------
__global__ void k_finalize(float* __restrict__ agg,
                           const float* __restrict__ H,
                           const float* __restrict__ dinv,
                           const float* __restrict__ bias) {
    int gid = blockIdx.x * blockDim.x + threadIdx.x;
    const int total = N_NODES_C * DIM;
    if (gid >= total) return;
    const int node = gid >> 6;
    const int f = gid & 63;
    const float di = dinv[node];
    float v = agg[gid] + H[gid] * di * di + bias[f];
    agg[gid] = v > 0.0f ? v : 0.0f;
}

// ---------------- global mean pool (accumulate) ----------------
__global__ void k_pool_accum(const float* __restrict__ h,
                             const int* __restrict__ batch,
                             float* __restrict__ pooled,
                             float* __restrict__ cnt) {
    int gid = blockIdx.x * blockDim.x + threadIdx.x;
    const int total = N_NODES_C * DIM;
    if (gid >= total) return;
    const int node = gid >> 6;
    const int f = gid & 63;
    const int g = batch[node];
    atomicAdd(&pooled[g * DIM + f], h[gid]);
    if (f == 0) atomicAdd(&cnt[g], 1.0f);
}

// ---------------- head: out = (pooled/cnt) @ Wfc + bfc ----------------
__global__ void k_head(const float* __restrict__ pooled,
                       const float* __restrict__ cnt,
                       const float* __restrict__ Wfc,
                       const float* __restrict__ bfc,
                       float* __restrict__ out) {
    int t = blockIdx.x * blockDim.x + threadIdx.x;
    if (t >= N_GRAPHS_C * OUTD) return;
    const int g = t >> 1;
    const int o = t & 1;
    float c = cnt[g];
    c = c > 1.0f ? c : 1.0f;
    const float inv = 1.0f / c;
    float s = 0.0f;
    #pragma unroll
    for (int k = 0; k < DIM; ++k)
        s += pooled[g * DIM + k] * inv * Wfc[k * OUTD + o];
    out[t] = s + bfc[o];
}

// ---------------- launcher ----------------
extern "C" void kernel_launch(void* const* d_in, const int* in_sizes, int n_in,
                              void* d_out, int out_size, void* d_ws, size_t ws_size,
                              hipStream_t stream) {
    const float* x    = (const float*)d_in[0];
    const int*   eix  = (const int*)d_in[1];
    const int*   srcp = eix;
    const int*   dstp = eix + N_EDGES_C;
    const int*   batch= (const int*)d_in[2];
    const float* W1   = (const float*)d_in[3];
    const float* b1   = (const float*)d_in[4];
    const float* W2   = (const float*)d_in[5];
    const float* b2   = (const float*)d_in[6];
    const float* Wfc  = (const float*)d_in[7];
    const float* bfc  = (const float*)d_in[8];
    float* out = (float*)d_out;

    char* ws = (char*)d_ws;
    float* dinv   = (float*)ws; ws += sizeof(float) * N_NODES_C;               // 400 KB (16B aligned)
    float* hbuf   = (float*)ws; ws += sizeof(float) * (size_t)N_NODES_C * DIM; // 25.6 MB
    float* agg    = (float*)ws; ws += sizeof(float) * (size_t)N_NODES_C * DIM; // 25.6 MB
    float* pooled = (float*)ws; ws += sizeof(float) * N_GRAPHS_C * DIM;
    float* cnt    = (float*)ws;

    const int NF = N_NODES_C * DIM;
    const int TB = 256;
    const int gNodes = (N_NODES_C + TB - 1) / TB;
    const int gEdges = (N_EDGES_C + TB - 1) / TB;
    const int gNF    = (NF + TB - 1) / TB;
    const int gScat  = (int)(((long long)N_EDGES_C * 16 + TB - 1) / TB);
    const int gGemm  = (N_NODES_C + 63) / 64;

    // degree / normalization
    k_init_deg<<<gNodes, TB, 0, stream>>>(dinv);
    k_accum_deg<<<gEdges, TB, 0, stream>>>(dstp, dinv);
    k_deg_to_dinv<<<gNodes, TB, 0, stream>>>(dinv);

    // ---- layer 1 ----
    k_zero_f32<<<gNF, TB, 0, stream>>>(agg, NF);
    k_gemm64_wmma<<<gGemm, 128, 0, stream>>>(x, W1, hbuf, N_NODES_C);
    k_scatter_edges<<<gScat, TB, 0, stream>>>(hbuf, srcp, dstp, dinv, agg);
    k_finalize<<<gNF, TB, 0, stream>>>(agg, hbuf, dinv, b1);   // agg <- relu(layer1)

    // ---- layer 2 ----
    k_gemm64_wmma<<<gGemm, 128, 0, stream>>>(agg, W2, hbuf, N_NODES_C);
    k_zero_f32<<<gNF, TB, 0, stream>>>(agg, NF);               // safe: after gemm read (stream order)
    k_scatter_edges<<<gScat, TB, 0, stream>>>(hbuf, srcp, dstp, dinv, agg);
    k_finalize<<<gNF, TB, 0, stream>>>(agg, hbuf, dinv, b2);   // agg <- relu(layer2)

    // ---- pool + head ----
    k_zero_f32<<<(N_GRAPHS_C * DIM + TB - 1) / TB, TB, 0, stream>>>(pooled, N_GRAPHS_C * DIM);
    k_zero_f32<<<(N_GRAPHS_C + TB - 1) / TB, TB, 0, stream>>>(cnt, N_GRAPHS_C);
    k_pool_accum<<<gNF, TB, 0, stream>>>(agg, batch, pooled, cnt);
    k_head<<<(N_GRAPHS_C * OUTD + TB - 1) / TB, TB, 0, stream>>>(pooled, cnt, Wfc, bfc, out);
}